// Loss_8143257993489
// MI455X (gfx1250) — compile-verified
//
#include <hip/hip_runtime.h>

// ---------------------------------------------------------------------------
// Focal/coord detection loss for MI455X (gfx1250, wave32).
// Memory-bound reduction (~13MB in, 3 floats out => ~0.6us at 23.3 TB/s).
// WMMA f32 16x16x4 used as the wave-level sum-reduction engine.
// ---------------------------------------------------------------------------

typedef __attribute__((ext_vector_type(2))) float v2f;
typedef __attribute__((ext_vector_type(8))) float v8f;

#define THRESH_F 0.5f
#define ALPHA_CLASS_F 0.01f
#define ALPHA_COORD_F 1.0f
#define MAXG 64

// ---- wave32 full sum via V_WMMA_F32_16X16X4_F32 ---------------------------
// A layout (32-bit 16x4): lanes 0-15 {VGPR0->K0, VGPR1->K1}, lanes 16-31
// {VGPR0->K2, VGPR1->K3}, M = lane%16.  With a = {v, 0} and B == all-ones,
// D[m][n] = v_m + v_{m+16}.  D VGPR j: lanes 0-15 hold M=j, lanes 16-31 hold
// M=j+8.  So sum_j d[j] gives half-sum; one shfl_xor(16) completes it.
__device__ __forceinline__ float wave_sum32(float v) {
#if __has_builtin(__builtin_amdgcn_wmma_f32_16x16x4_f32)
    v2f a; a[0] = v;    a[1] = 0.0f;
    v2f b; b[0] = 1.0f; b[1] = 1.0f;
    v8f c = {};
    c = __builtin_amdgcn_wmma_f32_16x16x4_f32(false, a, false, b,
                                              (short)0, c, false, false);
    float s = ((c[0] + c[1]) + (c[2] + c[3])) + ((c[4] + c[5]) + (c[6] + c[7]));
    s += __shfl_xor(s, 16, 32);
    return s;
#else
    float s = v;
    for (int off = 16; off > 0; off >>= 1) s += __shfl_xor(s, off, 32);
    return s;
#endif
}

// Deterministic block sum, broadcast to all threads. red[] must hold >=16 f32.
// NOTE: plain (non-volatile) LDS — __syncthreads() provides the ordering;
// volatile would force generic flat SYS-scope traffic on gfx1250.
__device__ __forceinline__ float block_reduce_sum(float v, float* red) {
    __syncthreads();                  // protect red[] across back-to-back calls
    float s = wave_sum32(v);
    const int lane = threadIdx.x & 31;
    const int wid  = threadIdx.x >> 5;
    if (lane == 0) red[wid] = s;
    __syncthreads();
    if (threadIdx.x == 0) {
        float t = 0.0f;
        const int nw = (blockDim.x + 31) >> 5;
        for (int i = 0; i < nw; ++i) t += red[i];
        red[15] = t;
    }
    __syncthreads();
    return red[15];
}

__device__ __forceinline__ float iou_pair(float ax0, float ay0, float ax1, float ay1,
                                          float areaA,
                                          float gx0, float gy0, float gx1, float gy1,
                                          float areaG) {
    float lx = fmaxf(ax0, gx0), ly = fmaxf(ay0, gy0);
    float rx = fminf(ax1, gx1), ry = fminf(ay1, gy1);
    float w = fmaxf(rx - lx, 0.0f), h = fmaxf(ry - ly, 0.0f);
    float inter = w * h;
    return inter / (areaA + areaG - inter);   // denom >= max(areaA,areaG) > 0
}

// orderable encoding: larger float => larger uint
__device__ __forceinline__ unsigned order_key(float f) {
    unsigned u = __float_as_uint(f);
    return (u & 0x80000000u) ? ~u : (u | 0x80000000u);
}

__global__ void initKeysKernel(unsigned long long* __restrict__ keys, int n) {
    for (int i = blockIdx.x * blockDim.x + threadIdx.x; i < n;
         i += gridDim.x * blockDim.x)
        keys[i] = 0ull;
}

// ---- pass 1: per (b,g) argmax_a(iou) --------------------------------------
// Each wave keeps its own per-g running max (shuffle-only, no block syncs in
// the loop); ONE barrier at the end, then threads g<G fold the 8 wave maxima
// and issue a single exact u64 atomicMax (deterministic).
__global__ void bestAnchorKernel(const float* __restrict__ anchors,  // [A,4] xyxy
                                 const float* __restrict__ gt,       // [B,G,4] xywh
                                 const int*   __restrict__ nobj,     // [B]
                                 unsigned long long* __restrict__ bestKeys, // [B,G]
                                 int A, int G) {
    __shared__ float4 sGT[MAXG];
    __shared__ float  sAreaG[MAXG];
    __shared__ int    sValid[MAXG];
    __shared__ unsigned long long sMax[8][MAXG];   // [wave][g]

    const int b   = blockIdx.y;
    const int tid = threadIdx.x;
    if (tid < G) {
        const float4 gw = ((const float4*)gt)[(size_t)b * G + tid]; // cx cy w h
        const float x0 = gw.x - 0.5f * gw.z, y0 = gw.y - 0.5f * gw.w;
        const float x1 = gw.x + 0.5f * gw.z, y1 = gw.y + 0.5f * gw.w;
        sGT[tid]    = make_float4(x0, y0, x1, y1);
        sAreaG[tid] = (x1 - x0) * (y1 - y0);
        sValid[tid] = (tid < nobj[b]) ? 1 : 0;
    }
    __syncthreads();

    const int a = blockIdx.x * blockDim.x + tid;
    float ax0 = 0.f, ay0 = 0.f, ax1 = 0.f, ay1 = 0.f, areaA = 0.f;
    const bool aok = (a < A);
    if (aok) {
        const float4 av = ((const float4*)anchors)[a];
        ax0 = av.x; ay0 = av.y; ax1 = av.z; ay1 = av.w;
        areaA = (ax1 - ax0) * (ay1 - ay0);
    }

    const int lane = tid & 31;
    const int wid  = tid >> 5;
    for (int g = 0; g < G; ++g) {
        if (!sValid[g]) continue;          // block-uniform branch
        unsigned long long key = 0ull;
        if (aok) {
            const float4 gb = sGT[g];
            const float iou = iou_pair(ax0, ay0, ax1, ay1, areaA,
                                       gb.x, gb.y, gb.z, gb.w, sAreaG[g]);
            // high bits: orderable iou; low bits: inverted index so ties pick
            // the SMALLEST anchor index (jnp.argmax first-occurrence rule)
            key = (((unsigned long long)order_key(iou)) << 32) |
                  (unsigned long long)(0xFFFFFFFFu - (unsigned)a);
        }
        for (int off = 16; off > 0; off >>= 1) {
            unsigned long long o = __shfl_xor(key, off, 32);
            key = (o > key) ? o : key;
        }
        if (lane == 0) sMax[wid][g] = key;
    }
    __syncthreads();

    if (tid < G && sValid[tid]) {
        unsigned long long m = sMax[0][tid];
        const int nw = blockDim.x >> 5;
        for (int i = 1; i < nw; ++i) m = (sMax[i][tid] > m) ? sMax[i][tid] : m;
        atomicMax(&bestKeys[(size_t)b * G + tid], m);
    }
}

// ---- pass 2: focal class loss + masked L1 coord partials ------------------
__global__ void lossKernel(const float* __restrict__ boxes,    // [B,A,4] raw xywh
                           const float* __restrict__ classes,  // [B,A,2]
                           const float* __restrict__ anchors,  // [A,4] xyxy
                           const float* __restrict__ gt,       // [B,G,4] xywh
                           const int*   __restrict__ nobj,     // [B]
                           const unsigned long long* __restrict__ bestKeys,
                           float* __restrict__ classP,
                           float* __restrict__ coordP,
                           float* __restrict__ npairP,
                           int A, int G) {
    __shared__ float4   sGT[MAXG];
    __shared__ float    sAreaG[MAXG];
    __shared__ int      sValid[MAXG];
    __shared__ unsigned sBest[MAXG];
    __shared__ float    red[16];

    const int b   = blockIdx.y;
    const int tid = threadIdx.x;
    if (tid < G) {
        const float4 gw = ((const float4*)gt)[(size_t)b * G + tid]; // cx cy w h
        const float x0 = gw.x - 0.5f * gw.z, y0 = gw.y - 0.5f * gw.w;
        const float x1 = gw.x + 0.5f * gw.z, y1 = gw.y + 0.5f * gw.w;
        sGT[tid]    = make_float4(x0, y0, x1, y1);
        sAreaG[tid] = (x1 - x0) * (y1 - y0);
        sValid[tid] = (tid < nobj[b]) ? 1 : 0;
        sBest[tid]  = 0xFFFFFFFFu -
                      (unsigned)(bestKeys[(size_t)b * G + tid] & 0xFFFFFFFFull);
    }
    __syncthreads();

    const int a = blockIdx.x * blockDim.x + tid;
    float cls = 0.0f, coord = 0.0f, npairs = 0.0f;
    if (a < A) {
        const float4 av = ((const float4*)anchors)[a];
        const float areaA = (av.z - av.x) * (av.w - av.y);

        const size_t ba = (size_t)b * A + a;
        const float4 bx = ((const float4*)boxes)[ba];
        const float2 cc = ((const float2*)classes)[ba];

        bool anyPos = false;
        for (int g = 0; g < G; ++g) {
            if (!sValid[g]) continue;
            const float4 gb = sGT[g];
            const float iou = iou_pair(av.x, av.y, av.z, av.w, areaA,
                                       gb.x, gb.y, gb.z, gb.w, sAreaG[g]);
            const bool pos = (iou > THRESH_F) || ((unsigned)a == sBest[g]);
            if (pos) {
                anyPos = true;
                coord += fabsf(bx.x - gb.x) + fabsf(bx.y - gb.y) +
                         fabsf(bx.z - gb.z) + fabsf(bx.w - gb.w);
                npairs += 1.0f;
            }
        }
        const float p  = anyPos ? cc.y : cc.x;
        const float om = 1.0f - p;
        cls = -om * om * logf(p);           // focal, gamma = 2
    }

    const float cT  = block_reduce_sum(cls,    red);
    const float coT = block_reduce_sum(coord,  red);
    const float nT  = block_reduce_sum(npairs, red);
    if (tid == 0) {
        const int idx = blockIdx.y * gridDim.x + blockIdx.x;  // b*NB + bx
        classP[idx] = cT;
        coordP[idx] = coT;
        npairP[idx] = nT;
    }
}

// ---- finalize: deterministic tree over the partials -----------------------
__global__ void finalizeKernel(const float* __restrict__ classP,
                               const float* __restrict__ coordP,
                               const float* __restrict__ npairP,
                               float* __restrict__ out, int NB, int B) {
    __shared__ float red[16];
    const int tid = threadIdx.x;

    float s = 0.0f;
    const int npt = NB * B;
    for (int i = tid; i < npt; i += blockDim.x) s += classP[i];
    const float classTot = block_reduce_sum(s, red);

    float coordAcc = 0.0f;     // identical on every thread (broadcast reduce)
    for (int b = 0; b < B; ++b) {
        float cs = 0.0f, ns = 0.0f;
        for (int i = tid; i < NB; i += blockDim.x) {
            cs += coordP[(size_t)b * NB + i];
            ns += npairP[(size_t)b * NB + i];
        }
        const float cb = block_reduce_sum(cs, red);
        const float nb = block_reduce_sum(ns, red);
        coordAcc += cb / (4.0f * nb);       // nb >= 1 (forced positives exist)
    }

    if (tid == 0) {
        const float class_loss = classTot * (ALPHA_CLASS_F / (float)B);
        const float coord_loss = coordAcc * (ALPHA_COORD_F / (float)B);
        out[0] = class_loss + coord_loss;   // total
        out[1] = class_loss;
        out[2] = coord_loss;
    }
}

extern "C" void kernel_launch(void* const* d_in, const int* in_sizes, int n_in,
                              void* d_out, int out_size, void* d_ws, size_t ws_size,
                              hipStream_t stream) {
    (void)n_in; (void)out_size; (void)ws_size;
    const float* boxes   = (const float*)d_in[0];  // [B,A,4]
    const float* classes = (const float*)d_in[1];  // [B,A,2]
    const float* anchors = (const float*)d_in[2];  // [A,4]
    const float* gt      = (const float*)d_in[3];  // [B,G,4]
    const int*   nobj    = (const int*)  d_in[4];  // [B]

    const int A = in_sizes[2] / 4;
    const int B = in_sizes[4];
    const int G = in_sizes[3] / (B * 4);

    const int BLOCK = 256;                  // 8 wave32 per block
    const int NB = (A + BLOCK - 1) / BLOCK;

    unsigned long long* bestKeys = (unsigned long long*)d_ws;
    float* classP = (float*)((char*)d_ws + sizeof(unsigned long long) * (size_t)B * G);
    float* coordP = classP + (size_t)B * NB;
    float* npairP = coordP + (size_t)B * NB;

    initKeysKernel<<<1, 256, 0, stream>>>(bestKeys, B * G);
    bestAnchorKernel<<<dim3(NB, B), BLOCK, 0, stream>>>(anchors, gt, nobj,
                                                        bestKeys, A, G);
    lossKernel<<<dim3(NB, B), BLOCK, 0, stream>>>(boxes, classes, anchors, gt,
                                                  nobj, bestKeys,
                                                  classP, coordP, npairP, A, G);
    finalizeKernel<<<1, 256, 0, stream>>>(classP, coordP, npairP,
                                          (float*)d_out, NB, B);
}